// GCN_71296457113906
// MI455X (gfx1250) — compile-verified
//
#include <hip/hip_runtime.h>

typedef float v2f __attribute__((ext_vector_type(2)));
typedef float v8f __attribute__((ext_vector_type(8)));

#define HDIM 128
#define LDSTRIDE 132  // 128 + 4 pad: column reads hit distinct LDS banks

static __device__ __forceinline__ void atomAdd(float* p, float v) {
    __hip_atomic_fetch_add(p, v, __ATOMIC_RELAXED, __HIP_MEMORY_SCOPE_AGENT);
}

// ---------------- normalization precompute ----------------

__global__ void k_init_deg(float* deg, int N) {
    int i = blockIdx.x * blockDim.x + threadIdx.x;
    if (i < N) deg[i] = 1.0f;  // self-loop contributes 1 to every node's degree
}

__global__ void k_edge_deg(const int* __restrict__ col, float* deg, int E) {
    int e = blockIdx.x * blockDim.x + threadIdx.x;
    if (e < E) atomAdd(&deg[col[e]], 1.0f);
}

__global__ void k_node_norm(float* __restrict__ deg_dinv,
                            const float* __restrict__ ncent,
                            float* __restrict__ lnorm, int N) {
    int i = blockIdx.x * blockDim.x + threadIdx.x;
    if (i < N) {
        float d  = deg_dinv[i];
        float di = (d > 0.0f) ? rsqrtf(d) : 0.0f;
        deg_dinv[i] = di;                  // in-place: deg -> deg^-1/2
        lnorm[i]    = di * di * ncent[i];  // self-loop edge norm
    }
}

__global__ void k_edge_norm(const int* __restrict__ row, const int* __restrict__ col,
                            const float* __restrict__ dinv,
                            const float* __restrict__ ecent,
                            float* __restrict__ enorm, int E) {
    int e = blockIdx.x * blockDim.x + threadIdx.x;
    if (e < E) enorm[e] = dinv[row[e]] * dinv[col[e]] * ecent[e];
}

// ---------------- f32 WMMA GEMM: Out[N,128] = X[N,128] @ W[128,128] ----------------
// 256 threads = 8 wave32s per block; block owns a 16-row slab of X (staged in LDS),
// wave w computes the 16x16 tile at columns [16w, 16w+16).

__global__ __launch_bounds__(256) void k_gemm_wmma(const float* __restrict__ X,
                                                   const float* __restrict__ W,
                                                   float* __restrict__ Out, int N) {
    __shared__ float As[16 * LDSTRIDE];
    const int row0 = blockIdx.x * 16;
    const int tid  = threadIdx.x;
    const int lane = tid & 31;
    const int wave = tid >> 5;

    int nrows = N - row0; if (nrows > 16) nrows = 16;
    const bool full = (nrows == 16);   // uniform across the block

    // Cooperative slab load: 16 rows x 32 float4 = 512 float4, 2 per thread.
    {
        const float4* src = (const float4*)(X + (size_t)row0 * HDIM);
#pragma unroll
        for (int i = 0; i < 2; ++i) {
            int idx = tid + i * 256;      // 0..511
            int r   = idx >> 5;           // row (32 float4 per row)
            int c   = idx & 31;           // float4 column
            float4 v = make_float4(0.f, 0.f, 0.f, 0.f);
            if (r < nrows) v = src[idx];
            ((float4*)(As + r * LDSTRIDE))[c] = v;
        }
    }
    __syncthreads();

    const int col0 = wave * 16;
    const int mn   = lane & 15;           // A: row index M; B: col index N
    const int kh   = (lane >> 4) << 1;    // lanes 0-15 -> K{0,1}; lanes 16-31 -> K{2,3}

    v8f acc = {};
#pragma unroll 8
    for (int k = 0; k < HDIM; k += 4) {
        v2f a, b;
        a[0] = As[mn * LDSTRIDE + k + kh];
        a[1] = As[mn * LDSTRIDE + k + kh + 1];
        b[0] = W[(size_t)(k + kh)     * HDIM + col0 + mn];
        b[1] = W[(size_t)(k + kh + 1) * HDIM + col0 + mn];
        acc = __builtin_amdgcn_wmma_f32_16x16x4_f32(false, a, false, b,
                                                    (short)0, acc, false, false);
    }

    // C layout: VGPR r -> M = r + (lane<16 ? 0 : 8); N = lane&15.
    const int cmBase = (lane >> 4) << 3;
    float* outp = Out + (size_t)(row0 + cmBase) * HDIM + col0 + mn;
    if (full) {
        // Fast path (uniform): 8 stores at immediate offsets, no per-lane guards.
#pragma unroll
        for (int r = 0; r < 8; ++r)
            outp[(size_t)r * HDIM] = acc[r];
    } else {
#pragma unroll
        for (int r = 0; r < 8; ++r) {
            if (row0 + cmBase + r < N)
                outp[(size_t)r * HDIM] = acc[r];
        }
    }
}

// ---------------- aggregation ----------------

// Self-loop term; also initializes Agg (one writer per element, no atomics).
__global__ __launch_bounds__(256) void k_selfloop(const float* __restrict__ lnorm,
                                                  const float* __restrict__ Hin,
                                                  float* __restrict__ Agg, int N) {
    int wid = blockIdx.x * 8 + (threadIdx.x >> 5);
    if (wid >= N) return;
    int lane = threadIdx.x & 31;
    float nrm = lnorm[wid];
    float4 v = ((const float4*)(Hin + (size_t)wid * HDIM))[lane];
    float4 o; o.x = nrm * v.x; o.y = nrm * v.y; o.z = nrm * v.z; o.w = nrm * v.w;
    ((float4*)(Agg + (size_t)wid * HDIM))[lane] = o;
}

// One wave per edge: gather h[row] (float4/lane), scatter-add into agg[col].
__global__ __launch_bounds__(256) void k_scatter(const int* __restrict__ row,
                                                 const int* __restrict__ col,
                                                 const float* __restrict__ enorm,
                                                 const float* __restrict__ Hin,
                                                 float* __restrict__ Agg, int E) {
    int e = blockIdx.x * 8 + (threadIdx.x >> 5);
    if (e >= E) return;
    int lane = threadIdx.x & 31;
    int r = row[e], c = col[e];
    float nrm = enorm[e];
    float4 v = ((const float4*)(Hin + (size_t)r * HDIM))[lane];
    float* p = Agg + (size_t)c * HDIM + lane * 4;
    atomAdd(p + 0, nrm * v.x);
    atomAdd(p + 1, nrm * v.y);
    atomAdd(p + 2, nrm * v.z);
    atomAdd(p + 3, nrm * v.w);
}

__global__ void k_relu_bias(const float* __restrict__ Agg,
                            const float* __restrict__ bias,
                            float* __restrict__ Xout, int total) {
    int i = blockIdx.x * blockDim.x + threadIdx.x;
    if (i < total) {
        float v = Agg[i] + bias[i & (HDIM - 1)];
        Xout[i] = v > 0.0f ? v : 0.0f;
    }
}

// ---------------- pooling + classifier ----------------

__global__ void k_zero(float* p, int n) {
    int i = blockIdx.x * blockDim.x + threadIdx.x;
    if (i < n) p[i] = 0.0f;
}

__global__ __launch_bounds__(256) void k_pool(const float* __restrict__ X,
                                              const int* __restrict__ batch,
                                              float* __restrict__ pooled,
                                              float* __restrict__ cnt, int N) {
    int wid = blockIdx.x * 8 + (threadIdx.x >> 5);
    if (wid >= N) return;
    int lane = threadIdx.x & 31;
    int g = batch[wid];
    float4 v = ((const float4*)(X + (size_t)wid * HDIM))[lane];
    float* p = pooled + (size_t)g * HDIM + lane * 4;
    atomAdd(p + 0, v.x);
    atomAdd(p + 1, v.y);
    atomAdd(p + 2, v.z);
    atomAdd(p + 3, v.w);
    if (lane == 0) atomAdd(&cnt[g], 1.0f);
}

__global__ void k_classify(const float* __restrict__ pooled,
                           const float* __restrict__ cnt,
                           const float* __restrict__ Wc,
                           const float* __restrict__ bc,
                           float* __restrict__ out, int G, int C) {
    int i = blockIdx.x * blockDim.x + threadIdx.x;
    if (i >= G * C) return;
    int g = i / C, c = i % C;
    float inv = 1.0f / fmaxf(cnt[g], 1.0f);
    float acc = bc[c];
    for (int f = 0; f < HDIM; ++f)
        acc += pooled[(size_t)g * HDIM + f] * inv * Wc[f * C + c];
    out[i] = acc;
}

// ---------------- launch ----------------

static inline int cdiv(int a, int b) { return (a + b - 1) / b; }

extern "C" void kernel_launch(void* const* d_in, const int* in_sizes, int n_in,
                              void* d_out, int out_size, void* d_ws, size_t ws_size,
                              hipStream_t stream) {
    const float* x     = (const float*)d_in[0];
    const int*   eidx  = (const int*)d_in[1];
    const int*   batch = (const int*)d_in[2];
    const float* ncent = (const float*)d_in[3];
    const float* ecent = (const float*)d_in[4];
    const float* W1 = (const float*)d_in[5];
    const float* b1 = (const float*)d_in[6];
    const float* W2 = (const float*)d_in[7];
    const float* b2 = (const float*)d_in[8];
    const float* W3 = (const float*)d_in[9];
    const float* b3 = (const float*)d_in[10];
    const float* Wc = (const float*)d_in[11];
    const float* bc = (const float*)d_in[12];
    float* out = (float*)d_out;

    const int N = in_sizes[3];       // node_centrality
    const int E = in_sizes[4];       // edge_centrality
    const int C = in_sizes[12];      // bc
    const int G = out_size / C;

    const int* row = eidx;           // edge_index[0]
    const int* col = eidx + E;       // edge_index[1]

    // workspace carve (floats)
    float* ws = (float*)d_ws;
    size_t off = 0;
    float* dinv   = ws + off; off += N;                  // degree -> deg^-1/2 in place
    float* lnorm  = ws + off; off += N;
    float* enorm  = ws + off; off += E;
    float* bufH   = ws + off; off += (size_t)N * HDIM;
    float* bufA   = ws + off; off += (size_t)N * HDIM;
    float* bufX   = ws + off; off += (size_t)N * HDIM;
    float* pooled = ws + off; off += (size_t)G * HDIM;
    float* cnt    = ws + off; off += G;                  // contiguous after pooled
    (void)ws_size; (void)n_in;

    dim3 b256(256);

    k_init_deg <<<cdiv(N, 256), b256, 0, stream>>>(dinv, N);
    k_edge_deg <<<cdiv(E, 256), b256, 0, stream>>>(col, dinv, E);
    k_node_norm<<<cdiv(N, 256), b256, 0, stream>>>(dinv, ncent, lnorm, N);
    k_edge_norm<<<cdiv(E, 256), b256, 0, stream>>>(row, col, dinv, ecent, enorm, E);

    const float* Ws[3] = {W1, W2, W3};
    const float* bs[3] = {b1, b2, b3};
    const float* xcur  = x;
    for (int l = 0; l < 3; ++l) {
        k_gemm_wmma<<<cdiv(N, 16), b256, 0, stream>>>(xcur, Ws[l], bufH, N);
        k_selfloop <<<cdiv(N, 8),  b256, 0, stream>>>(lnorm, bufH, bufA, N);
        k_scatter  <<<cdiv(E, 8),  b256, 0, stream>>>(row, col, enorm, bufH, bufA, E);
        k_relu_bias<<<cdiv(N * HDIM, 256), b256, 0, stream>>>(bufA, bs[l], bufX, N * HDIM);
        xcur = bufX;
    }

    k_zero    <<<cdiv(G * HDIM + G, 256), b256, 0, stream>>>(pooled, G * HDIM + G);
    k_pool    <<<cdiv(N, 8), b256, 0, stream>>>(bufX, batch, pooled, cnt, N);
    k_classify<<<cdiv(G * C, 256), b256, 0, stream>>>(pooled, cnt, Wc, bc, out, G, C);
}